// ParallelGatedRecurrence_17257178596088
// MI455X (gfx1250) — compile-verified
//
#include <hip/hip_runtime.h>
#include <math.h>

#define D_MODEL 1024
#define N_HEADS 8
#define D_HEAD  48
#define TOTAL   384
#define BATCH   4
#define SEQ     4096
#define M_TOTAL (BATCH*SEQ)   // 16384
#define N1      (3*TOTAL)     // 1152
#define NCHUNK  64
#define CHUNK   64            // SEQ / NCHUNK
#define GATE_LB 0.21176470588235294f
#define GN_EPS  1e-5f

typedef __attribute__((ext_vector_type(2))) float v2f;
typedef __attribute__((ext_vector_type(4))) float v4f;
typedef __attribute__((ext_vector_type(8))) float v8f;

__device__ __forceinline__ float sigm(float x) { return 1.0f / (1.0f + expf(-x)); }
__device__ __forceinline__ v2f lo2(v4f v) { return __builtin_shufflevector(v, v, 0, 1); }
__device__ __forceinline__ v2f hi2(v4f v) { return __builtin_shufflevector(v, v, 2, 3); }

#define WMMA_F32(a, b, c) \
    __builtin_amdgcn_wmma_f32_16x16x4_f32(false, (a), false, (b), (short)0, (c), false, false)

// ---------------------------------------------------------------------------
// Quantize the four weight matrices (ternary * scale). One block per matrix.
// Wq rows [0,384)=W_f, [384,768)=W_v, [768,1152)=W_o ; Wpq = W_proj quantized.
// ---------------------------------------------------------------------------
__global__ void pgr_quant(const float* __restrict__ Wf, const float* __restrict__ Wv,
                          const float* __restrict__ Wo, const float* __restrict__ Wp,
                          float* __restrict__ Wq, float* __restrict__ Wpq) {
    __shared__ float red[1024];
    const int which = blockIdx.x;
    const float* src = (which == 0) ? Wf : (which == 1) ? Wv : (which == 2) ? Wo : Wp;
    float* dst = (which == 3) ? Wpq : (Wq + which * TOTAL * D_MODEL);
    const int n = TOTAL * D_MODEL;   // 393216 elements for every matrix
    float s = 0.0f;
    for (int i = threadIdx.x; i < n; i += blockDim.x) s += fabsf(src[i]);
    red[threadIdx.x] = s;
    __syncthreads();
    for (int off = blockDim.x >> 1; off > 0; off >>= 1) {
        if (threadIdx.x < off) red[threadIdx.x] += red[threadIdx.x + off];
        __syncthreads();
    }
    float scale = fmaxf(red[0] / (float)n, 1e-5f);
    float inv = 1.0f / scale;
    for (int i = threadIdx.x; i < n; i += blockDim.x) {
        float q = rintf(src[i] * inv);            // round-half-even like jnp.round
        q = fminf(1.0f, fmaxf(-1.0f, q));
        dst[i] = q * scale;
    }
}

// ---------------------------------------------------------------------------
// f32 WMMA GEMM:  C(M x N) = A(M x K) @ Bw^T, Bw is (N x K) row-major.
// Wave computes a 16x64 tile via 4 accumulators; block = 4 waves = 64x64 tile.
// Macro K-step of 8: half-wave hi loads a contiguous float4 at column
// k + 4*hi (one b128), feeding TWO WMMAs:
//   WMMA#1 K-set {k,k+1,k+4,k+5}   -> operands (x,y) of a/b
//   WMMA#2 K-set {k+2,k+3,k+6,k+7} -> operands (z,w) of a/b
// A and B use the same position mapping, so every K is consumed exactly once.
// 5 b128 loads feed 8 WMMAs; next macro-step is prefetched before use.
// ---------------------------------------------------------------------------
__global__ void pgr_gemm_wmma(const float* __restrict__ A, const float* __restrict__ Bw,
                              float* __restrict__ C, int K, int ldc) {
    const int lane = threadIdx.x & 31;
    const int wave = threadIdx.x >> 5;
    const int lm = lane & 15;
    const int hi = lane >> 4;
    const int m0 = blockIdx.x * 64 + wave * 16;
    const int n0 = blockIdx.y * 64;

    const int aoff  = (m0 + lm) * K + 4 * hi;   // 16B aligned, < 2^25 elements
    const int boff  = (n0 + lm) * K + 4 * hi;
    const int bstep = 16 * K;

    // prologue: first macro-step's operands
    v4f a  = *(const v4f*)(A  + aoff);
    v4f b0 = *(const v4f*)(Bw + boff);
    v4f b1 = *(const v4f*)(Bw + boff + bstep);
    v4f b2 = *(const v4f*)(Bw + boff + 2 * bstep);
    v4f b3 = *(const v4f*)(Bw + boff + 3 * bstep);

    v8f c0 = {}, c1 = {}, c2 = {}, c3 = {};
    #pragma unroll 4
    for (int k = 8; k < K; k += 8) {
        // prefetch next macro-step before consuming current registers
        v4f an  = *(const v4f*)(A  + aoff + k);
        v4f bn0 = *(const v4f*)(Bw + boff + k);
        v4f bn1 = *(const v4f*)(Bw + boff + bstep + k);
        v4f bn2 = *(const v4f*)(Bw + boff + 2 * bstep + k);
        v4f bn3 = *(const v4f*)(Bw + boff + 3 * bstep + k);
        v2f al = lo2(a), ah = hi2(a);
        c0 = WMMA_F32(al, lo2(b0), c0);
        c1 = WMMA_F32(al, lo2(b1), c1);
        c2 = WMMA_F32(al, lo2(b2), c2);
        c3 = WMMA_F32(al, lo2(b3), c3);
        c0 = WMMA_F32(ah, hi2(b0), c0);
        c1 = WMMA_F32(ah, hi2(b1), c1);
        c2 = WMMA_F32(ah, hi2(b2), c2);
        c3 = WMMA_F32(ah, hi2(b3), c3);
        a = an; b0 = bn0; b1 = bn1; b2 = bn2; b3 = bn3;
    }
    // epilogue macro-step
    {
        v2f al = lo2(a), ah = hi2(a);
        c0 = WMMA_F32(al, lo2(b0), c0);
        c1 = WMMA_F32(al, lo2(b1), c1);
        c2 = WMMA_F32(al, lo2(b2), c2);
        c3 = WMMA_F32(al, lo2(b3), c3);
        c0 = WMMA_F32(ah, hi2(b0), c0);
        c1 = WMMA_F32(ah, hi2(b1), c1);
        c2 = WMMA_F32(ah, hi2(b2), c2);
        c3 = WMMA_F32(ah, hi2(b3), c3);
    }

    float* Cp = C + (size_t)(m0 + hi * 8) * ldc + n0 + lm;
    #pragma unroll
    for (int j = 0; j < 8; j++) {
        Cp[(size_t)j * ldc +  0] = c0[j];
        Cp[(size_t)j * ldc + 16] = c1[j];
        Cp[(size_t)j * ldc + 32] = c2[j];
        Cp[(size_t)j * ldc + 48] = c3[j];
    }
}

// ---------------------------------------------------------------------------
// Scan pass 1: per (b, chunk, channel) compute chunk gate-product G and
// chunk-local scan end value H (with h_in = 0).
// ---------------------------------------------------------------------------
__global__ void pgr_scan1(const float* __restrict__ Y, const float* __restrict__ fb,
                          float* __restrict__ cG, float* __restrict__ cH) {
    int idx = blockIdx.x * blockDim.x + threadIdx.x;     // [b][ch][c]
    if (idx >= BATCH * NCHUNK * TOTAL) return;
    int c  = idx % TOTAL;
    int ch = (idx / TOTAL) % NCHUNK;
    int b  = idx / (TOTAL * NCHUNK);
    const float* base = Y + (size_t)(b * SEQ + ch * CHUNK) * N1;
    float fbias = fb[c];
    float g = 1.0f, h = 0.0f;
    for (int t = 0; t < CHUNK; t++) {
        const float* row = base + (size_t)t * N1;
        float f = GATE_LB + (1.0f - GATE_LB) * sigm(row[c] + fbias);
        float v = row[TOTAL + c];
        h = f * h + (1.0f - f) * v;
        g *= f;
    }
    cG[idx] = g;
    cH[idx] = h;
}

// Scan pass 2: serial combine over the 64 chunks -> chunk-entry h (carry).
__global__ void pgr_scan2(const float* __restrict__ cG, const float* __restrict__ cH,
                          float* __restrict__ carry) {
    int idx = blockIdx.x * blockDim.x + threadIdx.x;     // B*TOTAL = 1536
    if (idx >= BATCH * TOTAL) return;
    int c = idx % TOTAL;
    int b = idx / TOTAL;
    float h = 0.0f;
    for (int ch = 0; ch < NCHUNK; ch++) {
        int o = (b * NCHUNK + ch) * TOTAL + c;
        carry[o] = h;
        h = cH[o] + cG[o] * h;
    }
}

// Scan pass 3: full scan with carry, apply out-gate, write gated hidden and
// deterministic per-(b,chunk,channel) partial sums for GroupNorm.
__global__ void pgr_scan3(const float* __restrict__ Y, const float* __restrict__ fb,
                          const float* __restrict__ carry, float* __restrict__ Hout,
                          float* __restrict__ pS, float* __restrict__ pQ) {
    int idx = blockIdx.x * blockDim.x + threadIdx.x;
    if (idx >= BATCH * NCHUNK * TOTAL) return;
    int c  = idx % TOTAL;
    int ch = (idx / TOTAL) % NCHUNK;
    int b  = idx / (TOTAL * NCHUNK);
    const float* base = Y + (size_t)(b * SEQ + ch * CHUNK) * N1;
    float* ob = Hout + (size_t)(b * SEQ + ch * CHUNK) * TOTAL + c;
    float fbias = fb[c];
    float h = carry[idx];
    float sum = 0.0f, sq = 0.0f;
    for (int t = 0; t < CHUNK; t++) {
        const float* row = base + (size_t)t * N1;
        float f  = GATE_LB + (1.0f - GATE_LB) * sigm(row[c] + fbias);
        float v  = row[TOTAL + c];
        float og = sigm(row[2 * TOTAL + c]);
        h = f * h + (1.0f - f) * v;
        float o = og * h;
        ob[(size_t)t * TOTAL] = o;
        sum += o;
        sq  += o * o;
    }
    pS[idx] = sum;
    pQ[idx] = sq;
}

// Deterministic GroupNorm statistics: one block per (b, head) group.
__global__ void pgr_gnstats(const float* __restrict__ pS, const float* __restrict__ pQ,
                            float* __restrict__ stats) {
    __shared__ float rs[256], rq[256];
    int g  = blockIdx.x;                 // 0..31
    int b  = g / N_HEADS;
    int hd = g % N_HEADS;
    float s = 0.0f, q = 0.0f;
    for (int i = threadIdx.x; i < NCHUNK * D_HEAD; i += blockDim.x) {
        int ch = i / D_HEAD;
        int cc = hd * D_HEAD + (i % D_HEAD);
        int o = (b * NCHUNK + ch) * TOTAL + cc;
        s += pS[o];
        q += pQ[o];
    }
    rs[threadIdx.x] = s; rq[threadIdx.x] = q;
    __syncthreads();
    for (int off = 128; off > 0; off >>= 1) {
        if (threadIdx.x < off) { rs[threadIdx.x] += rs[threadIdx.x + off]; rq[threadIdx.x] += rq[threadIdx.x + off]; }
        __syncthreads();
    }
    if (threadIdx.x == 0) {
        float n = (float)(SEQ * D_HEAD);
        float mean = rs[0] / n;
        float var  = rq[0] / n - mean * mean;
        stats[g]      = mean;
        stats[32 + g] = rsqrtf(var + GN_EPS);
    }
}

// Normalize in place: (x - mean) * rstd * gamma + beta.
__global__ void pgr_gnorm(float* __restrict__ Hout, const float* __restrict__ stats,
                          const float* __restrict__ gw, const float* __restrict__ gb) {
    size_t i = (size_t)blockIdx.x * blockDim.x + threadIdx.x;
    if (i >= (size_t)M_TOTAL * TOTAL) return;
    int c = (int)(i % TOTAL);
    int m = (int)(i / TOTAL);
    int b = m >> 12;                     // m / SEQ
    int g = b * N_HEADS + c / D_HEAD;
    float mean = stats[g];
    float rstd = stats[32 + g];
    Hout[i] = (Hout[i] - mean) * rstd * gw[c] + gb[c];
}

// ---------------------------------------------------------------------------
extern "C" void kernel_launch(void* const* d_in, const int* in_sizes, int n_in,
                              void* d_out, int out_size, void* d_ws, size_t ws_size,
                              hipStream_t stream) {
    const float* x   = (const float*)d_in[0];
    const float* Wf  = (const float*)d_in[1];
    const float* Wv  = (const float*)d_in[2];
    const float* Wo  = (const float*)d_in[3];
    const float* Wp  = (const float*)d_in[4];
    const float* fb  = (const float*)d_in[5];
    const float* gw  = (const float*)d_in[6];
    const float* gb  = (const float*)d_in[7];
    float* out = (float*)d_out;

    float* w     = (float*)d_ws;
    float* Wq    = w;                       // 1152*1024
    float* Wpq   = Wq   + N1 * D_MODEL;     // 1024*384
    float* Y     = Wpq  + D_MODEL * TOTAL;  // 16384*1152
    float* Hout  = Y    + (size_t)M_TOTAL * N1;     // 16384*384
    float* cG    = Hout + (size_t)M_TOTAL * TOTAL;  // 98304
    float* cH    = cG   + BATCH * NCHUNK * TOTAL;
    float* carry = cH   + BATCH * NCHUNK * TOTAL;
    float* pS    = carry + BATCH * NCHUNK * TOTAL;
    float* pQ    = pS   + BATCH * NCHUNK * TOTAL;
    float* stats = pQ   + BATCH * NCHUNK * TOTAL;   // 64 floats

    // 1) ternary-quantize all four weight matrices
    pgr_quant<<<4, 1024, 0, stream>>>(Wf, Wv, Wo, Wp, Wq, Wpq);

    // 2) fused pre-activation GEMM: Y = x @ [Wf|Wv|Wo]^T  (16384 x 1152)
    pgr_gemm_wmma<<<dim3(M_TOTAL / 64, N1 / 64), 128, 0, stream>>>(x, Wq, Y, D_MODEL, N1);

    // 3) chunked parallel scan
    int nthr = BATCH * NCHUNK * TOTAL;   // 98304
    pgr_scan1<<<(nthr + 255) / 256, 256, 0, stream>>>(Y, fb, cG, cH);
    pgr_scan2<<<(BATCH * TOTAL + 255) / 256, 256, 0, stream>>>(cG, cH, carry);
    pgr_scan3<<<(nthr + 255) / 256, 256, 0, stream>>>(Y, fb, carry, Hout, pS, pQ);

    // 4) GroupNorm (deterministic reduction), normalize in place
    pgr_gnstats<<<BATCH * N_HEADS, 256, 0, stream>>>(pS, pQ, stats);
    size_t nelem = (size_t)M_TOTAL * TOTAL;
    pgr_gnorm<<<(unsigned)((nelem + 255) / 256), 256, 0, stream>>>(Hout, stats, gw, gb);

    // 5) final projection: out = Hout @ Wproj^T  (16384 x 1024)
    pgr_gemm_wmma<<<dim3(M_TOTAL / 64, D_MODEL / 64), 128, 0, stream>>>(Hout, Wpq, out, TOTAL, D_MODEL);
}